// FarthestPointSampler_37812892074551
// MI455X (gfx1250) — compile-verified
//
#include <hip/hip_runtime.h>
#include <stdint.h>

// Problem constants (from the reference).
#define B_       32
#define N_       100000
#define NPTS     1024
#define NTHREADS 1024
#define NCHUNK   ((N_ + NTHREADS - 1) / NTHREADS)   // 98 chunks of 1024 points
#define KD       65000                               // distances resident in LDS
#define RING     4                                   // ring slots (per-thread 16B each)
#define DEPTH    3                                   // async prefetch depth

// Dynamic LDS layout:
//   [0, RING_BYTES)            : ring buffer, RING x 1024 x float4 (async landing zone)
//   [DIST_OFF, DIST_OFF+4*KD)  : resident per-point running distances
//   [RED_OFF ...)              : redV[32] | redI[32] | cen[4] | curFar
#define RING_BYTES (RING * NTHREADS * 16)            // 65536
#define DIST_OFF   RING_BYTES
#define RED_OFF    (DIST_OFF + 4 * KD)               // 325536
#define LDS_BYTES  (RED_OFF + 512)                   // 326048 <= 327680 (320KB)

// ---------------------------------------------------------------------------
// Kernel 1: pack coords as (x,y,z,|p|^2) float4 and init spilled distances.
// ---------------------------------------------------------------------------
__global__ void fps_pack_kernel(const float* __restrict__ coord,
                                float4* __restrict__ packed,
                                float* __restrict__ spill) {
  long long i = (long long)blockIdx.x * blockDim.x + threadIdx.x;
  const long long total = (long long)B_ * N_;
  if (i < total) {
    float x = coord[3 * i + 0];
    float y = coord[3 * i + 1];
    float z = coord[3 * i + 2];
    float4 p;
    p.x = x; p.y = y; p.z = z;
    p.w = x * x + y * y + z * z;
    packed[i] = p;
  }
  const long long spillTotal = (long long)B_ * (N_ - KD);
  if (i < spillTotal) spill[i] = 1e10f;
}

// ---------------------------------------------------------------------------
// Kernel 2: sequential FPS. One 1024-thread workgroup (32 wave32s) per batch.
// ---------------------------------------------------------------------------
__device__ __forceinline__ void async_b128(uint32_t lds_off, uint64_t gaddr) {
  // CDNA5 async copy: global -> LDS via the async engine (ASYNCcnt tracked).
  asm volatile("global_load_async_to_lds_b128 %0, %1, off"
               :: "v"(lds_off), "v"(gaddr) : "memory");
}

extern "C" __global__ void __launch_bounds__(NTHREADS)
fps_main_kernel(const float4* __restrict__ packed,
                const int* __restrict__ farthest_init,
                float* __restrict__ spill,
                float* __restrict__ out) {
  const int b   = blockIdx.x;
  const int tid = threadIdx.x;

  extern __shared__ unsigned char smem[];
  float* distL  = (float*)(smem + DIST_OFF);
  float* redV   = (float*)(smem + RED_OFF);
  int*   redI   = (int*)  (smem + RED_OFF + 128);
  float* cen    = (float*)(smem + RED_OFF + 256);   // cx, cy, cz, |c|^2
  int*   curFar = (int*)  (smem + RED_OFF + 272);

  const float4* pb = packed + (size_t)b * N_;
  float* sp        = spill  + (size_t)b * (N_ - KD);
  float* outIdx    = out    + (size_t)b * NPTS;
  float* outXYZ    = out + (size_t)B_ * NPTS + (size_t)b * NPTS * 3;

  const uint32_t ldsBase = (uint32_t)(uintptr_t)smem;   // ring at offset 0
  const uint64_t gBase   = (uint64_t)(uintptr_t)pb;

  // Init resident distance array; seed the first centroid index.
  for (int i = tid; i < KD; i += NTHREADS) distL[i] = 1e10f;
  if (tid == 0) *curFar = farthest_init[b];
  __syncthreads();

  for (int t = 0; t < NPTS; ++t) {
    // --- broadcast current centroid; record outputs (pre-update index) ---
    if (tid == 0) {
      const int f = *curFar;
      const float4 c = pb[f];                 // L2-hit gather
      cen[0] = c.x; cen[1] = c.y; cen[2] = c.z; cen[3] = c.w;
      outIdx[t] = (float)f;
      float* o = outXYZ + 3 * t;
      o[0] = c.x; o[1] = c.y; o[2] = c.z;
    }
    __syncthreads();
    const float cx = cen[0], cy = cen[1], cz = cen[2], cn = cen[3];

    float best  = -1.0f;
    int   besti = 0;

    // --- prime the async pipeline: chunks 0..DEPTH-1 ---
#pragma unroll
    for (int c = 0; c < DEPTH; ++c) {
      const int i = c * NTHREADS + tid;     // always < N_ for c < DEPTH
      const uint32_t l = ldsBase + (uint32_t)(((c & (RING - 1)) * NTHREADS + tid) * 16);
      async_b128(l, gBase + (uint64_t)i * 16u);
    }

    // --- stream all points: LDS ring + resident/spilled distance RMW ---
    for (int c = 0; c < NCHUNK; ++c) {
      const int i = c * NTHREADS + tid;
      if (c + DEPTH < NCHUNK)
        asm volatile("s_wait_asynccnt 0x2" ::: "memory");  // chunk c landed
      else
        asm volatile("s_wait_asynccnt 0x0" ::: "memory");  // drain tail
      if (i < N_) {
        const float4 p = ((const float4*)smem)[(c & (RING - 1)) * NTHREADS + tid];
        // prefetch chunk c+DEPTH into slot (c+DEPTH)&3 == (c-1)&3 (already consumed)
        if (c + DEPTH < NCHUNK) {
          const int j = (c + DEPTH) * NTHREADS + tid;
          if (j < N_) {
            const uint32_t l = ldsBase +
                (uint32_t)((((c + DEPTH) & (RING - 1)) * NTHREADS + tid) * 16);
            async_b128(l, gBase + (uint64_t)j * 16u);
          }
        }
        const float d = p.w - 2.0f * (p.x * cx + p.y * cy + p.z * cz) + cn;
        float nd;
        if (i < KD) {
          nd = fminf(distL[i], d);
          distL[i] = nd;
        } else {
          nd = fminf(sp[i - KD], d);
          sp[i - KD] = nd;
        }
        if (nd > best) { best = nd; besti = i; }   // first-max within thread
      }
    }

    // --- wave32 butterfly argmax (ties -> lower index, matching jnp.argmax) ---
#pragma unroll
    for (int m = 16; m > 0; m >>= 1) {
      const float ov = __shfl_xor(best, m, 32);
      const int   oi = __shfl_xor(besti, m, 32);
      if (ov > best || (ov == best && oi < besti)) { best = ov; besti = oi; }
    }
    if ((tid & 31) == 0) { redV[tid >> 5] = best; redI[tid >> 5] = besti; }
    __syncthreads();

    // --- 32 wave partials reduced by wave 0 ---
    if (tid < 32) {
      float v  = redV[tid];
      int   ix = redI[tid];
#pragma unroll
      for (int m = 16; m > 0; m >>= 1) {
        const float ov = __shfl_xor(v, m, 32);
        const int   oi = __shfl_xor(ix, m, 32);
        if (ov > v || (ov == v && oi < ix)) { v = ov; ix = oi; }
      }
      if (tid == 0) *curFar = ix;
    }
    __syncthreads();
  }
}

// ---------------------------------------------------------------------------
// Host launcher.
// ---------------------------------------------------------------------------
extern "C" void kernel_launch(void* const* d_in, const int* in_sizes, int n_in,
                              void* d_out, int out_size, void* d_ws, size_t ws_size,
                              hipStream_t stream) {
  (void)in_sizes; (void)n_in; (void)out_size; (void)ws_size;
  const float* coord = (const float*)d_in[0];
  const int*   finit = (const int*)d_in[1];
  float*       out   = (float*)d_out;

  // Workspace: [packed float4 B*N | spilled distances B*(N-KD)]
  float4* packed = (float4*)d_ws;
  float*  spill  = (float*)((char*)d_ws + (size_t)B_ * N_ * sizeof(float4));

  const long long total  = (long long)B_ * N_;
  const int       tpb    = 256;
  const int       blocks = (int)((total + tpb - 1) / tpb);
  fps_pack_kernel<<<blocks, tpb, 0, stream>>>(coord, packed, spill);
  fps_main_kernel<<<B_, NTHREADS, LDS_BYTES, stream>>>(packed, finit, spill, out);
}